// CombinedCache_30356828848372
// MI455X (gfx1250) — compile-verified
//
#include <hip/hip_runtime.h>

// Problem constants (B=1, H=32, Q=1024, S=4096, D=128)
#define H 32
#define Q 1024
#define S 4096
#define DD 128
#define NROWS (H * Q)          // 32768 rows to reduce over
#define START 4                // StreamingLLM sink tokens
#define HEAVY 409              // int(4096 * 0.1)
#define RECENT 409             // min(int(4096*0.1), 512)
#define RSTART (S - RECENT)    // 3687
#define MID_N (RSTART - START) // 3683 heavy-hitter candidates
#define KEEP_N (START + HEAVY + RECENT) // 822
#define NCHUNK 32
#define ROWS_PER_CHUNK (NROWS / NCHUNK) // 1024

typedef __attribute__((ext_vector_type(2))) float v2f;
typedef __attribute__((ext_vector_type(8))) float v8f;

// ---------------------------------------------------------------------------
// Kernel 1: column sums of attn_weights (32768 x 4096) via WMMA.
// colsum = ones(16x4) x W_tile(4x16), accumulated in the f32 16x16 C matrix.
// Each wave owns a 16-column tile and a 1024-row chunk; writes its partial
// sums to partial[chunk][col]. No atomics -> deterministic.
// B-matrix (4x16 f32) VGPR layout: v0 = rows K0 (lanes 0-15) / K2 (16-31),
//                                  v1 = rows K1 (lanes 0-15) / K3 (16-31).
// ---------------------------------------------------------------------------
__global__ __launch_bounds__(256) void colsum_wmma(
    const float* __restrict__ w, float* __restrict__ partial) {
  const int lane = threadIdx.x & 31;
  const int wave = blockIdx.x * (blockDim.x >> 5) + (threadIdx.x >> 5);
  const int colTile = wave & 255;   // 256 tiles of 16 columns
  const int chunk = wave >> 8;      // 32 row chunks
  const int col0 = colTile << 4;
  const int n = lane & 15;
  const int half = lane >> 4;       // 0 -> rows K0,K1 ; 1 -> rows K2,K3

  const float* base = w + (size_t)chunk * ROWS_PER_CHUNK * S + col0 + n;

  v2f a; a.x = 1.0f; a.y = 1.0f;    // ones matrix (A = 16x4)
  v8f c = {};                       // f32 accumulator

  for (int it = 0; it < ROWS_PER_CHUNK; it += 4) {
    const float* p = base + (size_t)(it + 2 * half) * S;
    v2f b;
    b.x = p[0];      // row K = 2*half
    b.y = p[S];      // row K = 2*half + 1
    c = __builtin_amdgcn_wmma_f32_16x16x4_f32(
        /*neg_a=*/false, a, /*neg_b=*/false, b,
        /*c_mod=*/(short)0, c, /*reuse_a=*/false, /*reuse_b=*/false);
  }
  // All 16 rows of D are identical column sums; lanes 0..15, VGPR0 = M=0,N=lane
  if (lane < 16) partial[chunk * S + col0 + n] = c[0];
}

// ---------------------------------------------------------------------------
// Kernel 2: combine the 32 partial-sum rows -> avg[4096] (sum; scale-invariant
// for top-k so no divide needed). Fixed order -> deterministic.
// ---------------------------------------------------------------------------
__global__ __launch_bounds__(256) void finalize_avg(
    const float* __restrict__ partial, float* __restrict__ avg) {
  int col = blockIdx.x * blockDim.x + threadIdx.x;
  if (col < S) {
    float s = 0.0f;
#pragma unroll
    for (int k = 0; k < NCHUNK; ++k) s += partial[k * S + col];
    avg[col] = s;
  }
}

// ---------------------------------------------------------------------------
// Kernel 3: exact rank-select over the middle region [START, RSTART).
// rank(i) = #{ j : v[j] > v[i]  ||  (v[j] == v[i] && j < i) }
// keep iff rank < HEAVY. Strict tie-break => ranks are a permutation =>
// exactly HEAVY survivors, same tie behavior as jax.lax.top_k.
// ---------------------------------------------------------------------------
__global__ __launch_bounds__(256) void rank_select(
    const float* __restrict__ avg, int* __restrict__ keepflag) {
  __shared__ float v[MID_N];
  for (int j = threadIdx.x; j < MID_N; j += blockDim.x) v[j] = avg[START + j];
  __syncthreads();
  int idx = blockIdx.x * blockDim.x + threadIdx.x;
  if (idx < MID_N) {
    const float vi = v[idx];
    int r = 0;
    for (int j = 0; j < MID_N; ++j) {
      float vj = v[j];
      r += (vj > vi) || (vj == vi && j < idx);
    }
    keepflag[START + idx] = (r < HEAVY) ? 1 : 0;
  }
}

// ---------------------------------------------------------------------------
// Kernel 4: build the sorted keep-index list (822 entries).
// keep[0..3] = 0..3 ; keep[4..412] = kept middle indices (ascending, via a
// segmented scan) ; keep[413..821] = 3687..4095.
// ---------------------------------------------------------------------------
__global__ __launch_bounds__(1024) void build_keep(
    const int* __restrict__ keepflag, int* __restrict__ keep) {
  __shared__ int flag[S];
  __shared__ int segbase[1024];
  const int t = threadIdx.x;
  for (int i = t; i < S; i += 1024)
    flag[i] = (i >= START && i < RSTART) ? keepflag[i] : 0;
  __syncthreads();

  int local = 0;
#pragma unroll
  for (int k = 0; k < 4; ++k) local += flag[t * 4 + k];
  segbase[t] = local;
  __syncthreads();

  if (t == 0) {  // serial exclusive scan over 1024 segment counts (tiny)
    int run = 0;
    for (int k = 0; k < 1024; ++k) { int cc = segbase[k]; segbase[k] = run; run += cc; }
  }
  __syncthreads();

  int run = segbase[t];
#pragma unroll
  for (int k = 0; k < 4; ++k) {
    int i = t * 4 + k;
    if (flag[i]) { keep[START + run] = i; ++run; }
  }
  if (t < START) keep[t] = t;                       // sink tokens
  if (t < RECENT) keep[START + HEAVY + t] = RSTART + t; // recent window
}

// ---------------------------------------------------------------------------
// Kernel 5: gather keys & values. One wave per 128-float output row,
// float4 (b128) loads/stores. Output = new_keys flat, then new_values flat.
// ---------------------------------------------------------------------------
__global__ __launch_bounds__(256) void gather_rows(
    const float* __restrict__ keys, const float* __restrict__ vals,
    const int* __restrict__ keep, float* __restrict__ out) {
  const int lane = threadIdx.x & 31;
  const int row = blockIdx.x * (blockDim.x >> 5) + (threadIdx.x >> 5);
  const int total = 2 * H * KEEP_N;
  if (row >= total) return;
  const int tsel = row / (H * KEEP_N);        // 0 = keys, 1 = values
  const int rem = row - tsel * (H * KEEP_N);
  const int h = rem / KEEP_N;
  const int o = rem - h * KEEP_N;
  const int src = keep[o];
  const float* in = (tsel ? vals : keys) + ((size_t)h * S + src) * DD;
  float* dst = out + ((size_t)tsel * H * KEEP_N + (size_t)h * KEEP_N + o) * DD;
  const float4* in4 = (const float4*)in;
  float4* dst4 = (float4*)dst;
  dst4[lane] = in4[lane];                     // 32 lanes x 16B = 512B/row
}

// ---------------------------------------------------------------------------
// Launch
// ---------------------------------------------------------------------------
extern "C" void kernel_launch(void* const* d_in, const int* in_sizes, int n_in,
                              void* d_out, int out_size, void* d_ws, size_t ws_size,
                              hipStream_t stream) {
  const float* keys = (const float*)d_in[0];   // (1,32,4096,128) f32
  const float* vals = (const float*)d_in[1];   // (1,32,4096,128) f32
  const float* attn = (const float*)d_in[2];   // (1,32,1024,4096) f32

  // Workspace layout (all written fresh every call):
  float* partial = (float*)d_ws;               // [32][4096] f32 = 512 KB
  float* avg = partial + NCHUNK * S;           // [4096] f32
  int* keepflag = (int*)(avg + S);             // [4096] i32
  int* keep = keepflag + S;                    // [1024] i32 (822 used)

  // 1) 537 MB column reduction via WMMA (8192 waves, 1024 blocks x 8 waves)
  colsum_wmma<<<(NCHUNK * 256) / 8, 256, 0, stream>>>(attn, partial);

  // 2) combine 32 partials per column
  finalize_avg<<<S / 256, 256, 0, stream>>>(partial, avg);

  // 3) exact top-k membership over the 3683 candidates
  rank_select<<<(MID_N + 255) / 256, 256, 0, stream>>>(avg, keepflag);

  // 4) sorted keep list
  build_keep<<<1, 1024, 0, stream>>>(keepflag, keep);

  // 5) gather: 2*32*822 rows, 8 rows (waves) per block
  gather_rows<<<(2 * H * KEEP_N + 7) / 8, 256, 0, stream>>>(
      keys, vals, keep, (float*)d_out);
}